// DecoderBlock_87737591922977
// MI455X (gfx1250) — compile-verified
//
#include <hip/hip_runtime.h>
#include <hip/hip_bf16.h>
#include <math.h>

#define HD 256
#define LN_EPS 1e-5f

typedef float v2f __attribute__((ext_vector_type(2)));
typedef float v8f __attribute__((ext_vector_type(8)));

// ---------------------------------------------------------------------------
// Small elementwise / edge kernels
// ---------------------------------------------------------------------------
__global__ void deg_kernel(const int* __restrict__ dst, const float* __restrict__ w,
                           float* __restrict__ deg, int E) {
    int e = blockIdx.x * blockDim.x + threadIdx.x;
    if (e < E) unsafeAtomicAdd(&deg[dst[e]], w[e]);
}

__global__ void dis_kernel(const float* __restrict__ deg, float* __restrict__ dis, int N) {
    int i = blockIdx.x * blockDim.x + threadIdx.x;
    if (i < N) {
        float d = deg[i];
        dis[i] = (d > 0.0f) ? rsqrtf(fmaxf(d, 1e-30f)) : 0.0f;
    }
}

__global__ void norm_kernel(const int* __restrict__ src, const int* __restrict__ dst,
                            const float* __restrict__ w, const float* __restrict__ dis,
                            float* __restrict__ nrm, int E) {
    int e = blockIdx.x * blockDim.x + threadIdx.x;
    if (e < E) nrm[e] = dis[src[e]] * w[e] * dis[dst[e]];
}

__global__ void add_kernel(const float* __restrict__ a, const float* __restrict__ b,
                           float* __restrict__ o, size_t n) {
    size_t i = (size_t)blockIdx.x * blockDim.x + threadIdx.x;
    if (i < n) o[i] = a[i] + b[i];
}

// ---------------------------------------------------------------------------
// GEMM: H[N,256] = A[N,256] @ W[256,256] using V_WMMA_F32_16X16X4_F32.
// Block = 256 threads = 8 waves; each wave computes a 16x256 row stripe.
// W K-panels (32x256 f32 = 32KB) staged in LDS *k-pair interleaved*:
//   sW2[(k/2)*256 + n] = { W[k][n], W[k+1][n] }   (8-byte unit)
// so each B fragment is one contiguous ds_load_b64 straight into a v2f.
// ---------------------------------------------------------------------------
__global__ __launch_bounds__(256) void gemm_kernel(const float* __restrict__ A,
                                                   const float* __restrict__ W,
                                                   float* __restrict__ H, int N) {
    __shared__ v2f sW2[16 * HD];           // 16 k-pairs x 256 cols = 32KB
    const int tid  = threadIdx.x;
    const int wave = tid >> 5;
    const int lane = tid & 31;
    const int l16  = lane & 15;
    const int hi   = lane >> 4;            // 0: lanes 0-15, 1: lanes 16-31
    const int rowBase = blockIdx.x * 128 + wave * 16;

    v8f acc[16];
#pragma unroll
    for (int t = 0; t < 16; ++t) {
#pragma unroll
        for (int r = 0; r < 8; ++r) acc[t][r] = 0.0f;
    }

    const float4* W4 = (const float4*)W;

    for (int kb = 0; kb < 8; ++kb) {       // 8 panels of 32 K each
        __syncthreads();
        // stage panel: 16 k-pairs x 64 float4-cols = 1024 groups, 4 per thread
#pragma unroll
        for (int j = tid; j < 1024; j += 256) {
            const int kp = j >> 6;          // k-pair within panel
            const int nq = j & 63;          // float4 column
            const float4 a4 = W4[(size_t)(kb * 32 + 2 * kp + 0) * 64 + nq];
            const float4 b4 = W4[(size_t)(kb * 32 + 2 * kp + 1) * 64 + nq];
            const int base = kp * HD + nq * 4;
            v2f p;
            p[0] = a4.x; p[1] = b4.x; sW2[base + 0] = p;
            p[0] = a4.y; p[1] = b4.y; sW2[base + 1] = p;
            p[0] = a4.z; p[1] = b4.z; sW2[base + 2] = p;
            p[0] = a4.w; p[1] = b4.w; sW2[base + 3] = p;
        }
        __syncthreads();

#pragma unroll
        for (int ks = 0; ks < 8; ++ks) {   // K steps of 4 inside the panel
            const int kloc = ks * 4 + 2 * hi;    // this half-wave's K offset
            const int kp   = ks * 2 + hi;        // its k-pair index
            // A fragment, ISA 16x4 layout: contiguous K pair -> one b64 load
            int row = rowBase + l16;
            if (row >= N) row = N - 1;     // clamp: keep EXEC all-ones for WMMA
            const float* ap = A + (size_t)row * HD + kb * 32 + kloc;
            v2f a;
            a[0] = ap[0];
            a[1] = ap[1];
            const v2f* bp = &sW2[kp * HD + l16];
#pragma unroll
            for (int nt = 0; nt < 16; ++nt) {
                const v2f b = bp[nt * 16];  // single contiguous 8B LDS load
                acc[nt] = __builtin_amdgcn_wmma_f32_16x16x4_f32(
                    false, a, false, b, (short)0, acc[nt], false, false);
            }
        }
    }

    // C/D layout: acc[nt][r] at lane l -> row rowBase + r + 8*hi, col nt*16 + l16
#pragma unroll
    for (int nt = 0; nt < 16; ++nt) {
        const int col = nt * 16 + l16;
#pragma unroll
        for (int r = 0; r < 8; ++r) {
            const int row = rowBase + r + 8 * hi;
            if (row < N) H[(size_t)row * HD + col] = acc[nt][r];
        }
    }
}

// ---------------------------------------------------------------------------
// Scatter-add: agg[dst[e]] += h[src[e]] * norm[e].  One wave per edge,
// 8 f32 global atomics per lane.
// ---------------------------------------------------------------------------
__global__ __launch_bounds__(256) void scatter_kernel(const float* __restrict__ h,
                                                      const int* __restrict__ src,
                                                      const int* __restrict__ dst,
                                                      const float* __restrict__ nrm,
                                                      float* __restrict__ agg, int E) {
    const int wave = threadIdx.x >> 5;
    const int lane = threadIdx.x & 31;
    const int e = blockIdx.x * 8 + wave;
    if (e >= E) return;
    const int s = src[e], d = dst[e];
    const float nv = nrm[e];
    const float4* hp = (const float4*)(h + (size_t)s * HD);
    float* ap = agg + (size_t)d * HD;
#pragma unroll
    for (int half = 0; half < 2; ++half) {
        const int j = lane + 32 * half;
        float4 v = hp[j];
        const int c = j * 4;
        unsafeAtomicAdd(ap + c + 0, v.x * nv);
        unsafeAtomicAdd(ap + c + 1, v.y * nv);
        unsafeAtomicAdd(ap + c + 2, v.z * nv);
        unsafeAtomicAdd(ap + c + 3, v.w * nv);
    }
}

// ---------------------------------------------------------------------------
// Fused bias + LayerNorm + exact GELU.  One wave (32 lanes) per 256-wide row.
// ---------------------------------------------------------------------------
__global__ __launch_bounds__(256) void ln_gelu_kernel(const float* __restrict__ in,
                                                      const float* __restrict__ bias,
                                                      const float* __restrict__ g,
                                                      const float* __restrict__ bln,
                                                      float* __restrict__ out, int N) {
    const int wave = threadIdx.x >> 5;
    const int lane = threadIdx.x & 31;
    const int row = blockIdx.x * 8 + wave;
    if (row >= N) return;
    const float* p = in + (size_t)row * HD;
    float v[8];
    float s = 0.0f;
#pragma unroll
    for (int j = 0; j < 8; ++j) {
        const int c = lane + 32 * j;
        v[j] = p[c] + bias[c];
        s += v[j];
    }
#pragma unroll
    for (int off = 16; off > 0; off >>= 1) s += __shfl_xor(s, off, 32);
    const float mean = s * (1.0f / HD);
    float q = 0.0f;
#pragma unroll
    for (int j = 0; j < 8; ++j) { const float d = v[j] - mean; q += d * d; }
#pragma unroll
    for (int off = 16; off > 0; off >>= 1) q += __shfl_xor(q, off, 32);
    const float rstd = rsqrtf(q * (1.0f / HD) + LN_EPS);
    float* o = out + (size_t)row * HD;
#pragma unroll
    for (int j = 0; j < 8; ++j) {
        const int c = lane + 32 * j;
        const float y = (v[j] - mean) * rstd * g[c] + bln[c];
        o[c] = 0.5f * y * (1.0f + erff(y * 0.70710678118654752f));
    }
}

// ---------------------------------------------------------------------------
// Final: io = (cur + io) / max(||cur+io||_2, 1e-8).  One wave per row;
// io holds the identity branch on entry and the result on exit (== d_out).
// ---------------------------------------------------------------------------
__global__ __launch_bounds__(256) void add_l2norm_kernel(const float* __restrict__ cur,
                                                         float* __restrict__ io, int N) {
    const int wave = threadIdx.x >> 5;
    const int lane = threadIdx.x & 31;
    const int row = blockIdx.x * 8 + wave;
    if (row >= N) return;
    const float* p = cur + (size_t)row * HD;
    float* o = io + (size_t)row * HD;
    float t[8];
    float q = 0.0f;
#pragma unroll
    for (int j = 0; j < 8; ++j) {
        const int c = lane + 32 * j;
        t[j] = p[c] + o[c];
        q += t[j] * t[j];
    }
#pragma unroll
    for (int off = 16; off > 0; off >>= 1) q += __shfl_xor(q, off, 32);
    const float inv = 1.0f / fmaxf(sqrtf(q), 1e-8f);
#pragma unroll
    for (int j = 0; j < 8; ++j) {
        const int c = lane + 32 * j;
        o[c] = t[j] * inv;
    }
}

// ---------------------------------------------------------------------------
extern "C" void kernel_launch(void* const* d_in, const int* in_sizes, int n_in,
                              void* d_out, int out_size, void* d_ws, size_t ws_size,
                              hipStream_t stream) {
    const float* x        = (const float*)d_in[0];
    const float* skip     = (const float*)d_in[1];
    const int*   ei       = (const int*)d_in[2];
    const float* ew       = (const float*)d_in[3];
    const float* conv_w   = (const float*)d_in[4];
    const float* conv_b   = (const float*)d_in[5];
    const float* ln_g     = (const float*)d_in[6];
    const float* ln_b     = (const float*)d_in[7];
    const float* res_w    = (const float*)d_in[8];
    const float* res_b    = (const float*)d_in[9];
    const float* res_ln_g = (const float*)d_in[10];
    const float* res_ln_b = (const float*)d_in[11];

    const int N = in_sizes[0] / HD;
    const int E = in_sizes[3];
    const int C = in_sizes[4] / (HD * HD);

    const size_t nd = (size_t)N * HD;
    float* ws  = (float*)d_ws;
    float* cur = ws;               // [N,256] current activations
    float* h   = cur + nd;         // [N,256] GEMM output / messages
    float* agg = h + nd;           // [N,256] scatter accumulator
    float* deg = agg + nd;         // [N]
    float* dis = deg + N;          // [N]
    float* nrm = dis + N;          // [E]

    const int* srcI = ei;          // edge_index[0]
    const int* dstI = ei + E;      // edge_index[1]

    float* outF = (float*)d_out;

    // ---- GCN symmetric normalization ----
    hipMemsetAsync(deg, 0, sizeof(float) * (size_t)N, stream);
    deg_kernel<<<(E + 255) / 256, 256, 0, stream>>>(dstI, ew, deg, E);
    dis_kernel<<<(N + 255) / 256, 256, 0, stream>>>(deg, dis, N);
    norm_kernel<<<(E + 255) / 256, 256, 0, stream>>>(srcI, dstI, ew, dis, nrm, E);

    // ---- combined = x + skip ----
    add_kernel<<<(int)((nd + 255) / 256), 256, 0, stream>>>(x, skip, cur, nd);

    // ---- residual branch: identity = gelu(LN(combined @ res_w + res_b)) -> d_out ----
    gemm_kernel<<<(N + 127) / 128, 256, 0, stream>>>(cur, res_w, h, N);
    ln_gelu_kernel<<<(N + 7) / 8, 256, 0, stream>>>(h, res_b, res_ln_g, res_ln_b, outF, N);

    // ---- conv stack ----
    for (int i = 0; i < C; ++i) {
        gemm_kernel<<<(N + 127) / 128, 256, 0, stream>>>(cur, conv_w + (size_t)i * HD * HD, h, N);
        hipMemsetAsync(agg, 0, sizeof(float) * nd, stream);
        scatter_kernel<<<(E + 7) / 8, 256, 0, stream>>>(h, srcI, dstI, nrm, agg, E);
        ln_gelu_kernel<<<(N + 7) / 8, 256, 0, stream>>>(agg, conv_b + (size_t)i * HD,
                                                        ln_g + (size_t)i * HD,
                                                        ln_b + (size_t)i * HD, cur, N);
    }

    // ---- out = L2-normalize(cur + identity) -> d_out ----
    add_l2norm_kernel<<<(N + 7) / 8, 256, 0, stream>>>(cur, outF, N);
}